// ModifiedSelfAttention_3281355014238
// MI455X (gfx1250) — compile-verified
//
#include <hip/hip_runtime.h>

// ---------------------------------------------------------------------------
// MI455X (gfx1250) wave32 WMMA implementation, full fp32 (V_WMMA_F32_16X16X4_F32)
// ---------------------------------------------------------------------------

#define DIM   1536
#define NH    12
#define HD    128
#define S_LEN 3744          // 3 * 26 * 48
#define MT    (S_LEN / 16)  // 234 row tiles

typedef __attribute__((ext_vector_type(2))) float v2f;
typedef __attribute__((ext_vector_type(8))) float v8f;

__device__ __forceinline__ v8f wmma_f32(v2f a, v2f b, v8f c) {
  // D = A(16x4) * B(4x16) + C(16x16), fp32
  return __builtin_amdgcn_wmma_f32_16x16x4_f32(
      /*neg_a=*/false, a, /*neg_b=*/false, b,
      /*c_mod=*/(short)0, c, /*reuse_a=*/false, /*reuse_b=*/false);
}

// ---------------------------------------------------------------------------
// GEMM: Y[M x 1536] = X[M x 1536] @ W^T + bias   (W is [1536 x 1536] row-major)
// block = 64 threads (2 waves); wave w handles M-tile (blockIdx.y*2 + w),
// N range blockIdx.x*64 .. +63 (4 subtiles of 16).
// ---------------------------------------------------------------------------
__global__ void gemm_bias_kernel(const float* __restrict__ X,
                                 const float* __restrict__ W,
                                 const float* __restrict__ bias,
                                 float* __restrict__ Y) {
  const int lane  = threadIdx.x & 31;
  const int wave  = threadIdx.x >> 5;
  const int mtile = blockIdx.y * 2 + wave;
  const int n0    = blockIdx.x * 64;
  const int lm    = lane & 15;
  const int g     = lane >> 4;

  const float* xrow = X + (size_t)(mtile * 16 + lm) * DIM;

  v8f acc[4];
#pragma unroll
  for (int j = 0; j < 4; ++j) acc[j] = (v8f){};

  for (int k = 0; k < DIM; k += 4) {
    const int kk = k + 2 * g;
    const v2f a = *(const v2f*)(xrow + kk);
#pragma unroll
    for (int j = 0; j < 4; ++j) {
      const float* wrow = W + (size_t)(n0 + j * 16 + lm) * DIM;
      const v2f b = *(const v2f*)(wrow + kk);
      acc[j] = wmma_f32(a, b, acc[j]);
    }
  }

#pragma unroll
  for (int j = 0; j < 4; ++j) {
    const int n  = n0 + j * 16 + lm;
    const float bv = bias[n];
#pragma unroll
    for (int r = 0; r < 8; ++r) {
      const int m = mtile * 16 + r + 8 * g;   // D layout: reg r -> row r + 8*(lane>>4)
      Y[(size_t)m * DIM + n] = acc[j][r] + bv;
    }
  }
}

// ---------------------------------------------------------------------------
// RMSNorm (over full DIM=1536) + 3D-grid RoPE, in place on Q or K.
// grid = (S_LEN, 2), block = 256.  Column split: 22 (f) | 21 (h) | 21 (w).
// ---------------------------------------------------------------------------
__global__ void norm_rope_kernel(float* __restrict__ Qb, float* __restrict__ Kb,
                                 const float* __restrict__ gq,
                                 const float* __restrict__ gk,
                                 const float* __restrict__ freqs) {
  const int s = blockIdx.x;
  float* row           = (blockIdx.y == 0 ? Qb : Kb) + (size_t)s * DIM;
  const float* gamma   = (blockIdx.y == 0 ? gq : gk);

  __shared__ float red[8];
  float ss = 0.f;
  for (int i = threadIdx.x; i < DIM; i += 256) {
    const float v = row[i];
    ss += v * v;
  }
#pragma unroll
  for (int off = 16; off >= 1; off >>= 1) ss += __shfl_xor(ss, off, 32);
  if ((threadIdx.x & 31) == 0) red[threadIdx.x >> 5] = ss;
  __syncthreads();
  float tot = 0.f;
#pragma unroll
  for (int i = 0; i < 8; ++i) tot += red[i];
  const float rinv = rsqrtf(tot * (1.0f / (float)DIM) + 1e-6f);

  const int w = s % 48;
  const int h = (s / 48) % 26;
  const int f = s / (48 * 26);

  // 12 heads * 64 complex pairs = 768 pairs per row
  for (int p = threadIdx.x; p < NH * 64; p += 256) {
    const int c  = p & 63;
    const int d  = (p >> 6) * HD + 2 * c;
    const int rr = (c < 22) ? f : ((c < 43) ? h : w);
    const float ang = freqs[rr * 64 + c];
    float sn, cs;
    __sincosf(ang, &sn, &cs);
    const float x0 = row[d]     * rinv * gamma[d];
    const float x1 = row[d + 1] * rinv * gamma[d + 1];
    row[d]     = x0 * cs - x1 * sn;
    row[d + 1] = x0 * sn + x1 * cs;
  }
}

// ---------------------------------------------------------------------------
// Flash attention, fp32 WMMA. One wave (32 threads) per (query-tile, head).
// Computes S^T = K_tile(16x128) x Q^T so softmax reduces over registers.
// P^T (C/D layout) -> A layout via shfl_xor(16) + half-selects (no LDS).
// ---------------------------------------------------------------------------
__global__ void attn_kernel(const float* __restrict__ Q,
                            const float* __restrict__ K,
                            const float* __restrict__ V,
                            float* __restrict__ O,
                            const int* __restrict__ seq_lens) {
  const int lane = threadIdx.x;
  const int qt   = blockIdx.x;   // 0..233
  const int h    = blockIdx.y;   // 0..11
  const int lm   = lane & 15;
  const int g    = lane >> 4;
  const int seqlen = seq_lens[0];
  const float scale = 0.088388347648318447f;  // 1/sqrt(128)

  // Preload Q^T fragments: B[kd][n=q] = Q[q][kd], 32 steps of K=4.
  v2f qf[32];
  const float* qrow = Q + (size_t)(qt * 16 + lm) * DIM + h * HD;
#pragma unroll
  for (int t = 0; t < 32; ++t) qf[t] = *(const v2f*)(qrow + 4 * t + 2 * g);

  v8f o[8];
#pragma unroll
  for (int ns = 0; ns < 8; ++ns) o[ns] = (v8f){};
  float m_i = -1e30f, l_i = 0.f;

  for (int kt = 0; kt < MT; ++kt) {
    // ---- S^T(16 keys x 16 queries) = K_tile x Q^T, K-dim = 128 ----
    v8f sacc = (v8f){};
    const float* krow = K + (size_t)(kt * 16 + lm) * DIM + h * HD;
#pragma unroll
    for (int t = 0; t < 32; ++t) {
      const v2f a = *(const v2f*)(krow + 4 * t + 2 * g);
      sacc = wmma_f32(a, qf[t], sacc);
    }

    // scale + mask; reg r holds key (kt*16 + r + 8*g), query column lm
    float sc[8];
#pragma unroll
    for (int r = 0; r < 8; ++r) {
      const int key = kt * 16 + r + 8 * g;
      sc[r] = (key < seqlen) ? sacc[r] * scale : -1e9f;
    }

    // ---- online softmax (per query column) ----
    float cmax = sc[0];
#pragma unroll
    for (int r = 1; r < 8; ++r) cmax = fmaxf(cmax, sc[r]);
    cmax = fmaxf(cmax, __shfl_xor(cmax, 16, 32));
    const float m_new = fmaxf(m_i, cmax);
    const float alpha = __expf(m_i - m_new);

    float p[8], rsum = 0.f;
#pragma unroll
    for (int r = 0; r < 8; ++r) {
      p[r] = __expf(sc[r] - m_new);
      rsum += p[r];
    }
    rsum += __shfl_xor(rsum, 16, 32);
    l_i = l_i * alpha + rsum;
    m_i = m_new;

    // rescale O: row m = r + 8*g needs alpha of query m (held at lane m)
    float alpham[8];
#pragma unroll
    for (int r = 0; r < 8; ++r) alpham[r] = __shfl(alpha, r + 8 * g, 32);
#pragma unroll
    for (int ns = 0; ns < 8; ++ns)
#pragma unroll
      for (int r = 0; r < 8; ++r) o[ns][r] *= alpham[r];

    // ---- relayout P^T (C/D layout) into A fragments (M=query, K=key) ----
    float sw[8];
#pragma unroll
    for (int r = 0; r < 8; ++r) sw[r] = __shfl_xor(p[r], 16, 32);
    v2f pa[4];
    pa[0].x = g ? sw[2] : p[0];  pa[0].y = g ? sw[3] : p[1];  // k = 0..3
    pa[1].x = g ? sw[6] : p[4];  pa[1].y = g ? sw[7] : p[5];  // k = 4..7
    pa[2].x = g ? p[2] : sw[0];  pa[2].y = g ? p[3] : sw[1];  // k = 8..11
    pa[3].x = g ? p[6] : sw[4];  pa[3].y = g ? p[7] : sw[5];  // k = 12..15

    // ---- O(16x128) += P(16x16) x V_tile(16x128) ----
    const float* vbase = V + (size_t)(kt * 16) * DIM + h * HD;
#pragma unroll
    for (int t = 0; t < 4; ++t) {
      const int k0 = 4 * t + 2 * g;  // B layout: reg.x -> K = 2*g within step
#pragma unroll
      for (int ns = 0; ns < 8; ++ns) {
        v2f b;
        b.x = vbase[(size_t)k0 * DIM + ns * 16 + lm];
        b.y = vbase[(size_t)(k0 + 1) * DIM + ns * 16 + lm];
        o[ns] = wmma_f32(pa[t], b, o[ns]);
      }
    }
  }

  // ---- epilogue: divide by l, store ----
  const float linv = 1.f / l_i;
  float linvm[8];
#pragma unroll
  for (int r = 0; r < 8; ++r) linvm[r] = __shfl(linv, r + 8 * g, 32);
#pragma unroll
  for (int ns = 0; ns < 8; ++ns) {
#pragma unroll
    for (int r = 0; r < 8; ++r) {
      const int m = qt * 16 + r + 8 * g;
      O[(size_t)m * DIM + h * HD + ns * 16 + lm] = o[ns][r] * linvm[r];
    }
  }
}

// ---------------------------------------------------------------------------
extern "C" void kernel_launch(void* const* d_in, const int* in_sizes, int n_in,
                              void* d_out, int out_size, void* d_ws, size_t ws_size,
                              hipStream_t stream) {
  (void)in_sizes; (void)n_in; (void)out_size; (void)ws_size;
  const float* x  = (const float*)d_in[0];
  const float* wq = (const float*)d_in[1];
  const float* bq = (const float*)d_in[2];
  const float* wk = (const float*)d_in[3];
  const float* bk = (const float*)d_in[4];
  const float* wv = (const float*)d_in[5];
  const float* bv = (const float*)d_in[6];
  const float* wo = (const float*)d_in[7];
  const float* bo = (const float*)d_in[8];
  const float* gq = (const float*)d_in[9];
  const float* gk = (const float*)d_in[10];
  const float* fr = (const float*)d_in[11];
  const int*   sl = (const int*)d_in[12];
  float* out = (float*)d_out;

  const size_t SD = (size_t)S_LEN * DIM;
  float* Qb = (float*)d_ws;
  float* Kb = Qb + SD;
  float* Vb = Kb + SD;
  float* Ab = Vb + SD;

  const dim3 gg(DIM / 64, MT / 2);  // (24, 117)
  const dim3 gb(64);

  gemm_bias_kernel<<<gg, gb, 0, stream>>>(x, wq, bq, Qb);
  gemm_bias_kernel<<<gg, gb, 0, stream>>>(x, wk, bk, Kb);
  gemm_bias_kernel<<<gg, gb, 0, stream>>>(x, wv, bv, Vb);

  norm_rope_kernel<<<dim3(S_LEN, 2), 256, 0, stream>>>(Qb, Kb, gq, gk, fr);

  attn_kernel<<<dim3(MT, NH), 32, 0, stream>>>(Qb, Kb, Vb, Ab, sl);

  gemm_bias_kernel<<<gg, gb, 0, stream>>>(Ab, wo, bo, out);
}